// AdditiveAttention_26319559590656
// MI455X (gfx1250) — compile-verified
//
#include <hip/hip_runtime.h>

// Problem constants (match reference)
#define B_  4
#define NQ_ 512
#define NK_ 512
#define D_  512
#define H_  128
#define QB_ 8     // query rows per scores-block

typedef __attribute__((ext_vector_type(2))) float v2f;
typedef __attribute__((ext_vector_type(8))) float v8f;

__device__ __forceinline__ float hw_tanh(float x) {
#if __has_builtin(__builtin_amdgcn_tanhf)
  return __builtin_amdgcn_tanhf(x);            // v_tanh_f32
#else
  float y;
  asm volatile("v_tanh_f32 %0, %1" : "=v"(y) : "v"(x));
  return y;
#endif
}

__device__ __forceinline__ float hw_exp(float x) {
#if __has_builtin(__builtin_amdgcn_exp2f)
  return __builtin_amdgcn_exp2f(x * 1.44269504088896340736f);  // v_exp_f32
#else
  return __expf(x);
#endif
}

// Fragment bundle for one K-step (K advances by 4): two A frags (M-tiles
// mTile, mTile+16) and four B frags (N-tiles nBase+0/16/32/48).
struct Frags {
  v2f a0, a1, b0, b1, b2, b3;
};

__device__ __forceinline__ void load_frags(const float* __restrict__ arow0,
                                           const float* __restrict__ arow1,
                                           const float* __restrict__ bbase,
                                           int N, int kk, Frags& f) {
  f.a0 = *(const v2f*)(arow0 + kk);            // A rows are 8B aligned (kHi even)
  f.a1 = *(const v2f*)(arow1 + kk);
  const float* b = bbase + (long long)kk * N;
  f.b0.x = b[0];       f.b0.y = b[N];
  f.b1.x = b[16];      f.b1.y = b[N + 16];
  f.b2.x = b[32];      f.b2.y = b[N + 32];
  f.b3.x = b[48];      f.b3.y = b[N + 48];
}

// -----------------------------------------------------------------------------
// Row-major f32 GEMM: C[M,N] = A[M,K] @ B[K,N] via V_WMMA_F32_16X16X4_F32.
// One wave per block computes a 32x64 C strip (2 M-tiles x 4 N-tiles,
// 8 accumulators). Software-pipelined: frags for step kk+4 are issued before
// the 8 WMMAs of step kk, so the load wait is covered by WMMA execution.
// grid = (M/32, N/64, batches).
//
// f32 WMMA fragment layouts (CDNA5 ISA 7.12.2):
//   A 16x4:  lane -> M = lane&15 ; VGPR j -> K = j + 2*(lane>>4)
//   B 4x16:  lane -> N = lane&15 ; VGPR j -> K = j + 2*(lane>>4)
//   C 16x16: lane -> N = lane&15 ; VGPR r -> M = r + 8*(lane>>4)
// -----------------------------------------------------------------------------
__global__ __launch_bounds__(32) void wmma_gemm_f32(
    const float* __restrict__ A, const float* __restrict__ Bm,
    float* __restrict__ C, int K, int N,
    long long sA, long long sB, long long sC) {
  const int lane  = threadIdx.x;
  const int mTile = blockIdx.x * 32;
  const int nBase = blockIdx.y * 64;
  const int batch = blockIdx.z;
  A  += (long long)batch * sA;
  Bm += (long long)batch * sB;
  C  += (long long)batch * sC;

  const int mRow = lane & 15;
  const int kHi  = (lane >> 4) * 2;   // lanes 16-31 hold K+2, K+3
  const int col  = lane & 15;

  const float* arow0 = A + (long long)(mTile + mRow) * K + kHi;
  const float* arow1 = arow0 + (long long)16 * K;
  const float* bbase = Bm + (long long)kHi * N + nBase + col;

  v8f acc00 = {}, acc01 = {}, acc02 = {}, acc03 = {};
  v8f acc10 = {}, acc11 = {}, acc12 = {}, acc13 = {};

  Frags cur, nxt;
  load_frags(arow0, arow1, bbase, N, 0, cur);

  for (int kk = 4; kk < K; kk += 4) {
    load_frags(arow0, arow1, bbase, N, kk, nxt);   // prefetch next step
    acc00 = __builtin_amdgcn_wmma_f32_16x16x4_f32(false, cur.a0, false, cur.b0, (short)0, acc00, false, false);
    acc01 = __builtin_amdgcn_wmma_f32_16x16x4_f32(false, cur.a0, false, cur.b1, (short)0, acc01, false, false);
    acc02 = __builtin_amdgcn_wmma_f32_16x16x4_f32(false, cur.a0, false, cur.b2, (short)0, acc02, false, false);
    acc03 = __builtin_amdgcn_wmma_f32_16x16x4_f32(false, cur.a0, false, cur.b3, (short)0, acc03, false, false);
    acc10 = __builtin_amdgcn_wmma_f32_16x16x4_f32(false, cur.a1, false, cur.b0, (short)0, acc10, false, false);
    acc11 = __builtin_amdgcn_wmma_f32_16x16x4_f32(false, cur.a1, false, cur.b1, (short)0, acc11, false, false);
    acc12 = __builtin_amdgcn_wmma_f32_16x16x4_f32(false, cur.a1, false, cur.b2, (short)0, acc12, false, false);
    acc13 = __builtin_amdgcn_wmma_f32_16x16x4_f32(false, cur.a1, false, cur.b3, (short)0, acc13, false, false);
    cur = nxt;
  }
  // epilogue step
  acc00 = __builtin_amdgcn_wmma_f32_16x16x4_f32(false, cur.a0, false, cur.b0, (short)0, acc00, false, false);
  acc01 = __builtin_amdgcn_wmma_f32_16x16x4_f32(false, cur.a0, false, cur.b1, (short)0, acc01, false, false);
  acc02 = __builtin_amdgcn_wmma_f32_16x16x4_f32(false, cur.a0, false, cur.b2, (short)0, acc02, false, false);
  acc03 = __builtin_amdgcn_wmma_f32_16x16x4_f32(false, cur.a0, false, cur.b3, (short)0, acc03, false, false);
  acc10 = __builtin_amdgcn_wmma_f32_16x16x4_f32(false, cur.a1, false, cur.b0, (short)0, acc10, false, false);
  acc11 = __builtin_amdgcn_wmma_f32_16x16x4_f32(false, cur.a1, false, cur.b1, (short)0, acc11, false, false);
  acc12 = __builtin_amdgcn_wmma_f32_16x16x4_f32(false, cur.a1, false, cur.b2, (short)0, acc12, false, false);
  acc13 = __builtin_amdgcn_wmma_f32_16x16x4_f32(false, cur.a1, false, cur.b3, (short)0, acc13, false, false);

  const int mOut0 = mTile + (lane >> 4) * 8;
#pragma unroll
  for (int r = 0; r < 8; ++r) {
    long long row0 = (long long)(mOut0 + r) * N + nBase + col;
    C[row0 +  0] = acc00[r];
    C[row0 + 16] = acc01[r];
    C[row0 + 32] = acc02[r];
    C[row0 + 48] = acc03[r];
    long long row1 = row0 + (long long)16 * N;
    C[row1 +  0] = acc10[r];
    C[row1 + 16] = acc11[r];
    C[row1 + 32] = acc12[r];
    C[row1 + 48] = acc13[r];
  }
}

// -----------------------------------------------------------------------------
// Fused scores + masked softmax. One 512-thread block handles QB_=8 query rows
// for one batch; thread k owns key k and streams its kproj row ONCE while
// accumulating 8 scores (8x reuse of kproj traffic). Softmax reductions use
// wave32 shfl_xor butterflies + one LDS combine stage (5 barriers total).
// Never materializes the [B,NQ,NK,H] intermediate (536 MB in the reference).
// -----------------------------------------------------------------------------
__global__ __launch_bounds__(NK_) void scores_softmax(
    const float* __restrict__ qproj, const float* __restrict__ kproj,
    const float* __restrict__ Wv, const int* __restrict__ valid_lens,
    float* __restrict__ attn) {
  __shared__ float qp[QB_][H_];
  __shared__ float wv[H_];
  __shared__ float part[QB_][16];
  __shared__ float mxs[QB_];
  __shared__ float dns[QB_];

  const int q0   = blockIdx.x * QB_;
  const int b    = blockIdx.y;
  const int k    = threadIdx.x;
  const int lane = k & 31;
  const int wid  = k >> 5;

  // cooperative loads (QB_*H_ = 1024 contiguous floats)
  for (int i = k; i < QB_ * H_; i += NK_)
    qp[i / H_][i % H_] = qproj[(long long)(b * NQ_ + q0) * H_ + i];
  if (k < H_) wv[k] = Wv[k];
  __syncthreads();

  const float4* kp4 =
      (const float4*)(kproj + (long long)(b * NK_ + k) * H_);
  float s[QB_];
#pragma unroll
  for (int j = 0; j < QB_; ++j) s[j] = 0.0f;

  for (int h4 = 0; h4 < H_ / 4; ++h4) {
    const float4 kv = kp4[h4];
    const int h = h4 * 4;
#pragma unroll
    for (int j = 0; j < QB_; ++j) {
      s[j] += wv[h + 0] * hw_tanh(qp[j][h + 0] + kv.x);
      s[j] += wv[h + 1] * hw_tanh(qp[j][h + 1] + kv.y);
      s[j] += wv[h + 2] * hw_tanh(qp[j][h + 2] + kv.z);
      s[j] += wv[h + 3] * hw_tanh(qp[j][h + 3] + kv.w);
    }
  }

  const int vlen = valid_lens[b];
  if (k >= vlen) {
#pragma unroll
    for (int j = 0; j < QB_; ++j) s[j] = -1.0e6f;
  }

  // ---- row max: wave butterfly + LDS combine ----
#pragma unroll
  for (int j = 0; j < QB_; ++j) {
    float m = s[j];
#pragma unroll
    for (int off = 16; off > 0; off >>= 1) m = fmaxf(m, __shfl_xor(m, off, 32));
    if (lane == 0) part[j][wid] = m;
  }
  __syncthreads();
  if (k < QB_) {
    float m = part[k][0];
#pragma unroll
    for (int w = 1; w < 16; ++w) m = fmaxf(m, part[k][w]);
    mxs[k] = m;
  }
  __syncthreads();

  // ---- exp + row sum ----
  float e[QB_];
#pragma unroll
  for (int j = 0; j < QB_; ++j) {
    e[j] = hw_exp(s[j] - mxs[j]);   // masked lanes underflow to exactly 0
    float t = e[j];
#pragma unroll
    for (int off = 16; off > 0; off >>= 1) t += __shfl_xor(t, off, 32);
    if (lane == 0) part[j][wid] = t;
  }
  __syncthreads();
  if (k < QB_) {
    float t = 0.0f;
#pragma unroll
    for (int w = 0; w < 16; ++w) t += part[k][w];
    dns[k] = t;
  }
  __syncthreads();

#pragma unroll
  for (int j = 0; j < QB_; ++j) {
    attn[(long long)(b * NQ_ + q0 + j) * NK_ + k] = e[j] / dns[j];
  }
}

// -----------------------------------------------------------------------------
// Host launcher
// Inputs: 0=queries [B,NQ,D] f32, 1=keys [B,NK,D] f32, 2=values [B,NK,D] f32,
//         3=Wq [D,H] f32, 4=Wk [D,H] f32, 5=Wv [H] f32, 6=valid_lens [B] i32
// Output: [B,NQ,DV] f32
// Workspace: qproj 1MB | kproj 1MB | attn 4MB  (6 MB total, L2-resident)
// -----------------------------------------------------------------------------
extern "C" void kernel_launch(void* const* d_in, const int* in_sizes, int n_in,
                              void* d_out, int out_size, void* d_ws, size_t ws_size,
                              hipStream_t stream) {
  (void)in_sizes; (void)n_in; (void)out_size; (void)ws_size;

  const float* queries    = (const float*)d_in[0];
  const float* keys       = (const float*)d_in[1];
  const float* values     = (const float*)d_in[2];
  const float* Wq         = (const float*)d_in[3];
  const float* Wk         = (const float*)d_in[4];
  const float* Wv         = (const float*)d_in[5];
  const int*   valid_lens = (const int*)d_in[6];
  float*       out        = (float*)d_out;

  float* qproj = (float*)d_ws;                        // B*NQ*H = 262144 f32
  float* kproj = qproj + (long long)B_ * NQ_ * H_;    // B*NK*H = 262144 f32
  float* attn  = kproj + (long long)B_ * NK_ * H_;    // B*NQ*NK = 1048576 f32

  // 1) qproj = queries @ Wq   (M = B*NQ = 2048, K = D, N = H)
  wmma_gemm_f32<<<dim3((B_ * NQ_) / 32, H_ / 64, 1), 32, 0, stream>>>(
      queries, Wq, qproj, D_, H_, 0, 0, 0);

  // 2) kproj = keys @ Wk
  wmma_gemm_f32<<<dim3((B_ * NK_) / 32, H_ / 64, 1), 32, 0, stream>>>(
      keys, Wk, kproj, D_, H_, 0, 0, 0);

  // 3) attn = masked_softmax( Wv . tanh(qproj + kproj) )
  scores_softmax<<<dim3(NQ_ / QB_, B_), NK_, 0, stream>>>(
      qproj, kproj, Wv, valid_lens, attn);

  // 4) out = attn @ values   (batched: M = NQ, K = NK, N = DV)
  wmma_gemm_f32<<<dim3(NQ_ / 32, D_ / 64, B_), 32, 0, stream>>>(
      attn, values, out, NK_, D_,
      (long long)NQ_ * NK_, (long long)NK_ * D_, (long long)NQ_ * D_);
}